// ZeroUpsample_24026047054687
// MI455X (gfx1250) — compile-verified
//
#include <hip/hip_runtime.h>

// Jittered 2x2 zero-upsample. Gather over output rows (cheap <=3-candidate
// scalar inversion), SCATTER over columns using a per-block precomputed
// forward column map in LDS (one fp32 map eval per input column per block).
// All HBM traffic dense: b32/b128 loads, async b128 LDS->global stores
// (CDNA5 ASYNCcnt path) with double-buffered row staging.

namespace {
constexpr int Bn = 8, Cn = 16;
constexpr int Hin = 540, Win = 960;
constexpr int Hout = 1080, Wout = 1920;
constexpr int RPB = 8;                         // output rows per block
constexpr int CHUNKS = Wout / 4;               // 480 float4 chunks per row

__device__ __forceinline__ int map_idx(int i, float jit, int hi) {
  // Bit-exact with reference: clip(floor(((i + 0.5) + jit) * 2.0), 0, hi)
  float v = ((float)i + 0.5f + jit) * 2.0f;
  int h = (int)floorf(v);
  h = h < 0 ? 0 : h;
  return h > hi ? hi : h;
}
}  // namespace

__global__ __launch_bounds__(256) void zero_upsample(
    const float* __restrict__ in, const float* __restrict__ jitter,
    float* __restrict__ out) {
  __shared__ int   s_wt[Win];                  // forward column map (per batch)
  __shared__ float s_row[2][Wout];             // double-buffered staged rows

  const int blk = blockIdx.x;                  // bc * (Hout/RPB) + rowchunk
  const int bc  = blk / (Hout / RPB);          // b*C + c
  const int H0  = (blk % (Hout / RPB)) * RPB;
  const int b   = bc / Cn;
  const int t   = threadIdx.x;

  const float jy = jitter[b * 2 + 0];
  const float jx = jitter[b * 2 + 1];

  // Forward column map: one eval per input column per block (no inversion).
  for (int x = t; x < Win; x += 256) s_wt[x] = map_idx(x, jx, Wout - 1);

  const float* src_base = in  + (size_t)bc * Hin  * Win;
  float*       dst_base = out + (size_t)bc * Hout * Wout;

  for (int r = 0; r < RPB; ++r) {
    const int H = H0 + r;
    float* buf = s_row[r & 1];

    // Buffer reuse: this wave's async stores from row r-2 (its oldest 2 ops,
    // in-order completion) must be done; barrier makes it true block-wide.
    if (r >= 2) asm volatile("s_wait_asynccnt 0x2" ::: "memory");
    __syncthreads();   // also covers s_wt availability on r == 0

    // Row inversion: largest y with clamp(floor((y+0.5+jy)*2)) == H.
    // Block-uniform -> SALU; <=3 candidates.
    int sy = -1;
    {
      int ylo = (H - 3) >> 1; if (ylo < 0) ylo = 0;
      int yhi = H >> 1;       if (yhi > Hin - 1) yhi = Hin - 1;
      for (int y = ylo; y <= yhi; ++y)
        if (map_idx(y, jy, Hout - 1) == H) sy = y;
    }

    // Zero the staged row (b128 LDS stores).
    {
      const float4 z = make_float4(0.f, 0.f, 0.f, 0.f);
      *(float4*)&buf[4 * t] = z;
      const int ci = t + 256;
      if (ci < CHUNKS) *(float4*)&buf[4 * ci] = z;
    }
    __syncthreads();

    // Scatter the source row: dense coalesced global b32 reads,
    // stride-1 LDS map reads, ~2-way-conflict LDS scatter writes.
    if (sy >= 0) {
      const float* src = src_base + (size_t)sy * Win;
      for (int x = t; x < Win; x += 256) {
        float v = src[x];
        buf[s_wt[x]] = v;
      }
    }
    __syncthreads();

    // Drain LDS -> HBM via CDNA5 async b128 stores. Chunk split {t, t+240}
    // with t<240: every wave (incl. partial wave 7) issues exactly 2 ops,
    // so the s_wait_asynccnt 2 double-buffer wait above is sound.
    float* dst = dst_base + (size_t)H * Wout;
    if (t < 240) {
      unsigned l0 = (unsigned)(unsigned long long)&buf[4 * t];
      unsigned l1 = (unsigned)(unsigned long long)&buf[4 * (t + 240)];
      float* g0 = dst + 4 * t;
      float* g1 = dst + 4 * (t + 240);
      asm volatile("global_store_async_from_lds_b128 %0, %1, off"
                   :: "v"(g0), "v"(l0) : "memory");
      asm volatile("global_store_async_from_lds_b128 %0, %1, off"
                   :: "v"(g1), "v"(l1) : "memory");
    }
  }

  asm volatile("s_wait_asynccnt 0x0" ::: "memory");
}

extern "C" void kernel_launch(void* const* d_in, const int* in_sizes, int n_in,
                              void* d_out, int out_size, void* d_ws, size_t ws_size,
                              hipStream_t stream) {
  const float* ten_in = (const float*)d_in[0];   // [8,16,540,960] f32
  const float* jitter = (const float*)d_in[1];   // [8,2,1,1] f32
  float* out = (float*)d_out;                    // [8,16,1080,1920] f32

  dim3 grid(Bn * Cn * (Hout / RPB));             // 128 * 135 = 17280 blocks
  zero_upsample<<<grid, 256, 0, stream>>>(ten_in, jitter, out);

  (void)in_sizes; (void)n_in; (void)out_size; (void)d_ws; (void)ws_size;
}